// Net_49641232007490
// MI455X (gfx1250) — compile-verified
//
#include <hip/hip_runtime.h>
#include <math.h>

// Problem constants (match reference).
constexpr int N_NODES = 2048;
constexpr int F_IN    = 2048;
constexpr int N_EDGES = 65536;
constexpr int G1      = 10;
constexpr int NCLS    = 10;
constexpr int WCOLS   = 32;   // padded [W0 | pad | W1 | pad] width (two 16-wide WMMA N-tiles)

typedef __attribute__((ext_vector_type(2))) float v2f;
typedef __attribute__((ext_vector_type(8))) float v8f;

// ---------------------------------------------------------------------------
// K0: zero the scratch regions we accumulate into (deg, y, agg).
__global__ void k_zero(float* __restrict__ p, int n) {
    for (int i = blockIdx.x * blockDim.x + threadIdx.x; i < n;
         i += gridDim.x * blockDim.x)
        p[i] = 0.0f;
}

// K1: deg[row[e]] += 1
__global__ void k_degree(const int* __restrict__ row, float* __restrict__ deg) {
    int e = blockIdx.x * blockDim.x + threadIdx.x;
    if (e < N_EDGES) atomicAdd(&deg[row[e]], 1.0f);
}

// K2: deg -> dis (in place). deg is integral, so max(deg,1)==deg when deg>0.
__global__ void k_rsqrt(float* __restrict__ deg) {
    int i = blockIdx.x * blockDim.x + threadIdx.x;
    if (i < N_NODES) {
        float d = deg[i];
        deg[i] = (d > 0.0f) ? rsqrtf(d) : 0.0f;
    }
}

// K3: pack W0/W1 into a WMMA-ready B layout.
// Logical B = [F_IN x 32]: cols 0..9 = W0, cols 16..25 = W1, zeros elsewhere.
// Physical:  Wpk[p*64 + n*2 + q] = B[K = 2p+q, n]   (p = K/2, q = K&1)
// so a lane's (B[kk,n], B[kk+1,n]) pair is one contiguous b64 load.
__global__ void k_pack_w(const float* __restrict__ W0, const float* __restrict__ W1,
                         float* __restrict__ Wpk) {
    int t = blockIdx.x * blockDim.x + threadIdx.x;
    if (t < F_IN * WCOLS) {
        int k = t >> 5, n = t & 31;          // K index, column
        float v = 0.0f;
        if (n < G1)                    v = W0[k * G1 + n];
        else if (n >= 16 && n < 16+G1) v = W1[k * G1 + (n - 16)];
        Wpk[(size_t)(k >> 1) * 64 + n * 2 + (k & 1)] = v;
    }
}

// K4: y[2048,32] += x[2048,2048] @ B[2048,32] using V_WMMA_F32_16X16X4_F32.
// One wave computes a 16x32 output tile over a 512-wide K chunk (KSPLIT=4),
// reducing into y via f32 atomics. Uniform control flow -> EXEC all ones.
// Inner loop VMEM: 1x b64 (A pair) + 2x b64 (B pairs) per K-step of 4.
__global__ void k_gemm_wmma(const float* __restrict__ x,
                            const float* __restrict__ Wpk,
                            float* __restrict__ y) {
    const int gw    = blockIdx.x * (blockDim.x >> 5) + (threadIdx.x >> 5);
    const int mTile = gw >> 2;         // 0..127
    const int kChnk = gw & 3;          // 0..3
    const int lane  = threadIdx.x & 31;
    const int lm    = lane & 15;       // M index (A) / N index (B,C)
    const int kh    = lane >> 4;       // K-half selector (0: K=0,1 ; 1: K=2,3)

    const float* __restrict__ xrow = x + (size_t)(mTile * 16 + lm) * F_IN;

    v8f c0 = {};   // N-tile 0 (cols 0..15  -> x@W0 in cols 0..9)
    v8f c1 = {};   // N-tile 1 (cols 16..31 -> x@W1 in cols 16..25)

    const int k0 = kChnk * 512;
#pragma unroll 8
    for (int k = k0; k < k0 + 512; k += 4) {
        const int kk = k + 2 * kh;     // this lane's first K value
        const int p  = kk >> 1;        // packed K-pair index
        // A 16x4 f32: lane lm holds A[M=lm, K=kk], A[M=lm, K=kk+1]
        v2f a  = *(const v2f*)(xrow + kk);
        // B 4x16 f32: lane lm holds B[K=kk, N], B[K=kk+1, N] (contiguous pair)
        v2f b0 = *(const v2f*)(Wpk + (size_t)p * 64 + lm * 2);        // N = lm
        v2f b1 = *(const v2f*)(Wpk + (size_t)p * 64 + 32 + lm * 2);   // N = 16+lm
        c0 = __builtin_amdgcn_wmma_f32_16x16x4_f32(false, a, false, b0,
                                                   (short)0, c0, false, false);
        c1 = __builtin_amdgcn_wmma_f32_16x16x4_f32(false, a, false, b1,
                                                   (short)0, c1, false, false);
    }

    // C/D layout: VGPR r holds M = r + 8*kh (lanes 16-31 are the +8 rows), N = lm.
    const int rowBase = mTile * 16 + 8 * kh;
#pragma unroll
    for (int r = 0; r < 8; ++r) {
        atomicAdd(&y[(size_t)(rowBase + r) * WCOLS + lm],      c0[r]);
        atomicAdd(&y[(size_t)(rowBase + r) * WCOLS + 16 + lm], c1[r]);
    }
}

// K5: edge aggregation in projected (10-dim) space:
//     agg[col, :] += w(e) * y1[row, :],  w(e) = -dis[row]*dis[col]
__global__ void k_scatter(const int* __restrict__ row, const int* __restrict__ col,
                          const float* __restrict__ dis, const float* __restrict__ y,
                          float* __restrict__ agg) {
    int e = blockIdx.x * blockDim.x + threadIdx.x;
    if (e < N_EDGES) {
        const int r = row[e], c = col[e];
        const float w = -dis[r] * dis[c];
        const float* __restrict__ yr = y + (size_t)r * WCOLS + 16;  // x@W1 slot
        float* __restrict__ ag = agg + (size_t)c * G1;
#pragma unroll
        for (int j = 0; j < G1; ++j) atomicAdd(&ag[j], w * yr[j]);
    }
}

// K6: h = relu(y0 + agg + b); logits = h @ Wf + bf; out = log_softmax(logits).
__global__ void k_head(const float* __restrict__ y, const float* __restrict__ agg,
                       const float* __restrict__ b, const float* __restrict__ Wf,
                       const float* __restrict__ bf, float* __restrict__ out) {
    int i = blockIdx.x * blockDim.x + threadIdx.x;
    if (i >= N_NODES) return;
    float h[G1];
#pragma unroll
    for (int j = 0; j < G1; ++j) {
        float v = y[(size_t)i * WCOLS + j] + agg[(size_t)i * G1 + j] + b[j];
        h[j] = v > 0.0f ? v : 0.0f;
    }
    float lg[NCLS];
    float m = -1e30f;
#pragma unroll
    for (int c = 0; c < NCLS; ++c) {
        float s = bf[c];
#pragma unroll
        for (int j = 0; j < G1; ++j) s = fmaf(h[j], Wf[j * NCLS + c], s);
        lg[c] = s;
        m = s > m ? s : m;
    }
    float se = 0.0f;
#pragma unroll
    for (int c = 0; c < NCLS; ++c) se += expf(lg[c] - m);
    const float lse = m + logf(se);
#pragma unroll
    for (int c = 0; c < NCLS; ++c) out[(size_t)i * NCLS + c] = lg[c] - lse;
}

// ---------------------------------------------------------------------------
extern "C" void kernel_launch(void* const* d_in, const int* in_sizes, int n_in,
                              void* d_out, int out_size, void* d_ws, size_t ws_size,
                              hipStream_t stream) {
    (void)in_sizes; (void)n_in; (void)out_size; (void)ws_size;

    const float* x   = (const float*)d_in[0];
    const int*   ei  = (const int*)  d_in[1];   // [2, E]: row = ei, col = ei + E
    const float* W0  = (const float*)d_in[2];
    const float* W1  = (const float*)d_in[3];
    const float* b   = (const float*)d_in[4];
    const float* Wf  = (const float*)d_in[5];
    const float* bf  = (const float*)d_in[6];
    float*       out = (float*)d_out;

    const int* row = ei;
    const int* col = ei + N_EDGES;

    // Workspace layout (floats): [deg/dis | y | agg | Wpk]
    float* ws   = (float*)d_ws;
    float* deg  = ws;                                   // 2048      (becomes dis)
    float* y    = deg + N_NODES;                        // 2048*32
    float* agg  = y   + N_NODES * WCOLS;                // 2048*10
    float* Wpk  = agg + N_NODES * G1;                   // 2048*32 (packed B)
    const int nZero = N_NODES + N_NODES * WCOLS + N_NODES * G1;  // contiguous deg+y+agg

    k_zero   <<<256, 256, 0, stream>>>(deg, nZero);
    k_degree <<<(N_EDGES + 255) / 256, 256, 0, stream>>>(row, deg);
    k_rsqrt  <<<(N_NODES + 255) / 256, 256, 0, stream>>>(deg);
    k_pack_w <<<(F_IN * WCOLS + 255) / 256, 256, 0, stream>>>(W0, W1, Wpk);
    // 128 M-tiles * 4 K-chunks = 512 waves; 8 waves/block -> 64 blocks.
    k_gemm_wmma<<<64, 256, 0, stream>>>(x, Wpk, y);
    k_scatter<<<(N_EDGES + 255) / 256, 256, 0, stream>>>(row, col, deg, y, agg);
    k_head   <<<(N_NODES + 255) / 256, 256, 0, stream>>>(y, agg, b, Wf, bf, out);
}